// MultiheadSelfAttention_74431783239836
// MI455X (gfx1250) — compile-verified
//
#include <hip/hip_runtime.h>
#include <stdint.h>
#include <math.h>

// ---------------------------------------------------------------------------
// MI455X (gfx1250) multi-head self-attention, fp32 WMMA + Tensor Data Mover.
// ---------------------------------------------------------------------------

typedef __attribute__((ext_vector_type(2))) float v2f;
typedef __attribute__((ext_vector_type(8))) float v8f;
typedef __attribute__((ext_vector_type(4))) unsigned int v4u;
typedef __attribute__((ext_vector_type(4))) int v4i;
typedef __attribute__((ext_vector_type(8))) int v8i;

#define WMMA4(A, B, C) \
    __builtin_amdgcn_wmma_f32_16x16x4_f32(false, (A), false, (B), (short)0, (C), false, false)

constexpr int Bn = 2;
constexpr int Sn = 2048;
constexpr int Cn = 2048;
constexpr int HD = 64;
constexpr int Hn = Cn / HD;          // 32
constexpr int Mn = Bn * Sn;          // 4096
constexpr int NQKV = 3 * Cn;         // 6144

#define NEG_INF (-3.402823466e38f)

// Full-wave (32 lane) sum reduction.
__device__ __forceinline__ float wsum32(float x) {
    x += __shfl_xor(x, 1, 32);
    x += __shfl_xor(x, 2, 32);
    x += __shfl_xor(x, 4, 32);
    x += __shfl_xor(x, 8, 32);
    x += __shfl_xor(x, 16, 32);
    return x;
}

// Byte offset of a generic pointer into LDS (low 32 bits of the flat address
// are the LDS byte offset per the CDNA5 aperture mapping).
__device__ __forceinline__ uint32_t lds_off(const void* p) {
    return (uint32_t)(uintptr_t)p;
}

// ---------------------------------------------------------------------------
// Tensor Data Mover: DMA a 2D tile (tile_d0 x tile_d1, fp32, row-contiguous in
// dim0) from a row-major global tensor (dim0 stride = d0_stride elements)
// into contiguous LDS at lds_byte.  D# per CDNA5 ISA ch.8 (CDNA5 ISA):
//   group0: [1:0]=count=1, [63:32]=lds_addr, [120:64]=global_addr, [127:126]=2
//   group1: [17:16]=data_size(2=4B), [79:48]=tensor_dim0, [111:80]=tensor_dim1,
//           [127:112]=tile_dim0, [143:128]=tile_dim1, [207:160]=dim0_stride
// ---------------------------------------------------------------------------
__device__ __forceinline__ void tdm_load_2d(const float* gptr, uint32_t lds_byte,
                                            uint32_t tensor_d0, uint32_t tensor_d1,
                                            uint32_t tile_d0, uint32_t tile_d1,
                                            uint32_t d0_stride) {
    const uint64_t ga = (uint64_t)(uintptr_t)gptr;
    v4u g0;
    g0[0] = 1u;                                            // count = 1, no gather
    g0[1] = lds_byte;                                      // lds_addr
    g0[2] = (uint32_t)ga;                                  // global_addr[31:0]
    g0[3] = (uint32_t)((ga >> 32) & 0x1FFFFFFu) | (2u << 30);  // [56:32] | type=2
    v8i g1;
    g1[0] = (int)(2u << 16);                               // data_size = 4 bytes
    g1[1] = (int)((tensor_d0 & 0xFFFFu) << 16);            // tensor_dim0[15:0]
    g1[2] = (int)((tensor_d0 >> 16) | ((tensor_d1 & 0xFFFFu) << 16));
    g1[3] = (int)((tensor_d1 >> 16) | (tile_d0 << 16));    // tile_dim0
    g1[4] = (int)(tile_d1 & 0xFFFFu);                      // tile_dim1 (tile_dim2=0)
    g1[5] = (int)d0_stride;                                // dim0 stride (elems)
    g1[6] = 0;
    g1[7] = 0;
    v4i z4 = {0, 0, 0, 0};
#if __clang_major__ >= 23
    v8i z8 = {0, 0, 0, 0, 0, 0, 0, 0};
    __builtin_amdgcn_tensor_load_to_lds(g0, g1, z4, z4, z8, 0);
#else
    __builtin_amdgcn_tensor_load_to_lds(g0, g1, z4, z4, 0);
#endif
}

// ---------------------------------------------------------------------------
// Kernel 1/4: Out[m,n] = sum_k A[m,k] * W[n,k]  (A: MxK, W: NxK, row-major).
// 128-thread block (4 waves), 64x64 block tile, 32-deep K chunks.
// TDM double-buffers A/B tiles in LDS; wave w computes rows [16w,16w+16) x 64.
// fp32 WMMA 16x16x4 fragment layout:
//   A frag: lane l<16 -> (M=l, K=j), lane>=16 -> (M=l&15, K=j+2)  (j = vgpr)
//   B frag: lane l<16 -> (K=j, N=l), lane>=16 -> (K=j+2, N=l&15)
//   C frag: vgpr r: lane l<16 -> (M=r, N=l), lane>=16 -> (M=r+8, N=l&15)
// ---------------------------------------------------------------------------
constexpr int BM = 64, BN = 64, KT = 32;

__global__ __launch_bounds__(128)
void gemm_abt_tdm(const float* __restrict__ A, const float* __restrict__ W,
                  float* __restrict__ Out, int M, int N, int K) {
    __shared__ float As[2][BM * KT];       // [row][k], 8 KB per buffer
    __shared__ float Bs[2][BN * KT];       // [col][k], 8 KB per buffer

    const int tid  = threadIdx.x;
    const int wid  = tid >> 5;
    const int lane = tid & 31;
    const int l    = lane & 15;
    const int half = lane >> 4;
    const int kb   = half * 2;
    const int m0   = blockIdx.x * BM;
    const int n0   = blockIdx.y * BN;
    const int nchunks = K / KT;

    // Chunk kc of A rows [m0,m0+64) and W rows [n0,n0+64) -> LDS buffer `buf`.
    auto issue = [&](int buf, int kc) {
        tdm_load_2d(A + (size_t)m0 * K + kc * KT, lds_off(&As[buf][0]),
                    (uint32_t)K, (uint32_t)M, KT, BM, (uint32_t)K);
        tdm_load_2d(W + (size_t)n0 * K + kc * KT, lds_off(&Bs[buf][0]),
                    (uint32_t)K, (uint32_t)N, KT, BN, (uint32_t)K);
    };

    if (wid == 0) {
        issue(0, 0);
        __builtin_amdgcn_s_wait_tensorcnt(0);
    }
    __syncthreads();

    v8f acc0 = {}, acc1 = {}, acc2 = {}, acc3 = {};
    int buf = 0;
    for (int kc = 0; kc < nchunks; ++kc) {
        if (kc + 1 < nchunks && wid == 0) issue(buf ^ 1, kc + 1);

        const float* as = &As[buf][(16 * wid + l) * KT];
        const float* b0 = &Bs[buf][(0 * 16 + l) * KT];
        const float* b1 = &Bs[buf][(1 * 16 + l) * KT];
        const float* b2 = &Bs[buf][(2 * 16 + l) * KT];
        const float* b3 = &Bs[buf][(3 * 16 + l) * KT];

#pragma unroll
        for (int t = 0; t < KT / 4; ++t) {
            v2f a  = *(const v2f*)(as + 4 * t + kb);
            v2f x0 = *(const v2f*)(b0 + 4 * t + kb);
            v2f x1 = *(const v2f*)(b1 + 4 * t + kb);
            v2f x2 = *(const v2f*)(b2 + 4 * t + kb);
            v2f x3 = *(const v2f*)(b3 + 4 * t + kb);
            acc0 = WMMA4(a, x0, acc0);
            acc1 = WMMA4(a, x1, acc1);
            acc2 = WMMA4(a, x2, acc2);
            acc3 = WMMA4(a, x3, acc3);
        }

        if (kc + 1 < nchunks) {
            if (wid == 0) __builtin_amdgcn_s_wait_tensorcnt(0);
            __syncthreads();
            buf ^= 1;
        }
    }

#pragma unroll
    for (int r = 0; r < 8; ++r) {
        const size_t row = (size_t)(m0 + 16 * wid + half * 8 + r) * N + n0;
        Out[row + 0 * 16 + l] = acc0[r];
        Out[row + 1 * 16 + l] = acc1[r];
        Out[row + 2 * 16 + l] = acc2[r];
        Out[row + 3 * 16 + l] = acc3[r];
    }
}

// ---------------------------------------------------------------------------
// Kernel 2: per-head LayerNorm (q,k) + rotary (q,k), repack q/k/v to [B,H,S,64].
// One wave per (b,s,h); lane i owns the rotary pair (2i, 2i+1).
// qkv layout: [B*S, 3C], o = h*192 + {q:0..63, k:64..127, v:128..191}
// rope layout: [S,1,32,2,2] -> flat s*128 + i*4 + p*2 + c
// ---------------------------------------------------------------------------
__global__ __launch_bounds__(32)
void ln_rope_pack(const float* __restrict__ qkv, const float* __restrict__ rope,
                  const float* __restrict__ qg, const float* __restrict__ qb,
                  const float* __restrict__ kg, const float* __restrict__ kb,
                  float* __restrict__ qp, float* __restrict__ kp,
                  float* __restrict__ vp) {
    const int idx = blockIdx.x;                  // ((b*S)+s)*H + h
    const int h = idx % Hn;
    const int s = (idx / Hn) % Sn;
    const int b = idx / (Hn * Sn);
    const int lane = threadIdx.x;                // pair index i
    const int e0 = 2 * lane, e1 = 2 * lane + 1;

    const size_t row = (size_t)(b * Sn + s) * NQKV + h * 192;
    float q0 = qkv[row + e0],       q1 = qkv[row + e1];
    float k0 = qkv[row + 64 + e0],  k1 = qkv[row + 64 + e1];
    float v0 = qkv[row + 128 + e0], v1 = qkv[row + 128 + e1];

    // LayerNorm(q) over 64 elements, biased variance, eps = 1e-5
    float mq = wsum32(q0 + q1) * (1.0f / 64.0f);
    float vq = wsum32((q0 - mq) * (q0 - mq) + (q1 - mq) * (q1 - mq)) * (1.0f / 64.0f);
    float rq = rsqrtf(vq + 1e-5f);
    q0 = (q0 - mq) * rq * qg[e0] + qb[e0];
    q1 = (q1 - mq) * rq * qg[e1] + qb[e1];

    // LayerNorm(k)
    float mk = wsum32(k0 + k1) * (1.0f / 64.0f);
    float vk = wsum32((k0 - mk) * (k0 - mk) + (k1 - mk) * (k1 - mk)) * (1.0f / 64.0f);
    float rk = rsqrtf(vk + 1e-5f);
    k0 = (k0 - mk) * rk * kg[e0] + kb[e0];
    k1 = (k1 - mk) * rk * kg[e1] + kb[e1];

    // Rotary: out[2i+p] = rope[s,0,i,p,0]*x[2i] + rope[s,0,i,p,1]*x[2i+1]
    const float* rp = rope + (size_t)s * 128 + lane * 4;
    const float r00 = rp[0], r01 = rp[1], r10 = rp[2], r11 = rp[3];
    const float qo0 = r00 * q0 + r01 * q1, qo1 = r10 * q0 + r11 * q1;
    const float ko0 = r00 * k0 + r01 * k1, ko1 = r10 * k0 + r11 * k1;

    const size_t ob = ((size_t)(b * Hn + h) * Sn + s) * HD + e0;
    qp[ob] = qo0; qp[ob + 1] = qo1;
    kp[ob] = ko0; kp[ob + 1] = ko1;
    vp[ob] = v0;  vp[ob + 1] = v1;
}

// ---------------------------------------------------------------------------
// Kernel 3: flash attention. One wave handles 16 query rows of one (b,h).
// Q tile (16x64) in registers as 16 A-fragments. Per 16-key block:
//   batch-load 16 K-fragments -> 16 score WMMAs (single load wait),
//   batch-prefetch 16 V-fragments, then online softmax (shfl row reductions,
//   overlapping the in-flight V loads), P through LDS into A-fragment layout,
//   16 P@V WMMAs from registers.
// q/k/v packed [B*H, S, 64]; out written as [B, S, C] (c = h*64 + d).
// ---------------------------------------------------------------------------
__global__ __launch_bounds__(32)
void flash_attn(const float* __restrict__ qp, const float* __restrict__ kp,
                const float* __restrict__ vp, float* __restrict__ out) {
    __shared__ float psh[16][17];                // 16x16 P tile (+pad)

    const int lane = threadIdx.x;
    const int l    = lane & 15;
    const int half = lane >> 4;
    const int kbo  = half * 2;
    const int q0   = blockIdx.x * 16;
    const int bh   = blockIdx.y;                 // b*H + h
    const int b    = bh / Hn;
    const int h    = bh % Hn;
    const size_t base = (size_t)bh * Sn * HD;

    // Q rows q0..q0+15 as A fragments (K-chunks t = 0..15 cover d = 0..63)
    v2f qa[16];
    const float* qrow = qp + base + (size_t)(q0 + l) * HD;
#pragma unroll
    for (int t = 0; t < 16; ++t)
        qa[t] = *(const v2f*)(qrow + 4 * t + kbo);

    float rmax[8], rsum[8], rscale[8];
    v8f oacc[4] = {};
#pragma unroll
    for (int r = 0; r < 8; ++r) { rmax[r] = NEG_INF; rsum[r] = 0.0f; }

    for (int j0 = 0; j0 < Sn; j0 += 16) {
        // ---- batch K-fragment loads, then the 16 score WMMAs ----
        v2f kf[16];
        const float* krow = kp + base + (size_t)(j0 + l) * HD;
#pragma unroll
        for (int t = 0; t < 16; ++t)
            kf[t] = *(const v2f*)(krow + 4 * t + kbo);

        v8f sc = {};
#pragma unroll
        for (int t = 0; t < 16; ++t)
            sc = WMMA4(qa[t], kf[t], sc);

        // ---- prefetch V fragments; latency hidden under softmax math ----
        v2f vf[4][4];
#pragma unroll
        for (int t = 0; t < 4; ++t) {
            const float* vr0 = vp + base + (size_t)(j0 + 4 * t + kbo) * HD;
            const float* vr1 = vr0 + HD;
#pragma unroll
            for (int nt = 0; nt < 4; ++nt) {
                vf[t][nt].x = vr0[nt * 16 + l];
                vf[t][nt].y = vr1[nt * 16 + l];
            }
        }

        // ---- online softmax; lane's half owns rows (half*8 .. half*8+7) ----
#pragma unroll
        for (int r = 0; r < 8; ++r) {
            float s = sc[r] * 0.125f;
            float m = s;
            m = fmaxf(m, __shfl_xor(m, 1, 32));
            m = fmaxf(m, __shfl_xor(m, 2, 32));
            m = fmaxf(m, __shfl_xor(m, 4, 32));
            m = fmaxf(m, __shfl_xor(m, 8, 32));
            const float mnew = fmaxf(rmax[r], m);
            const float p = __expf(s - mnew);
            float ps = p;
            ps += __shfl_xor(ps, 1, 32);
            ps += __shfl_xor(ps, 2, 32);
            ps += __shfl_xor(ps, 4, 32);
            ps += __shfl_xor(ps, 8, 32);
            rscale[r] = __expf(rmax[r] - mnew);
            rsum[r] = rsum[r] * rscale[r] + ps;
            rmax[r] = mnew;
            psh[r + half * 8][l] = p;            // psh[row][key]
        }
        __syncthreads();

        // rescale running output
#pragma unroll
        for (int nt = 0; nt < 4; ++nt)
#pragma unroll
            for (int r = 0; r < 8; ++r)
                oacc[nt][r] *= rscale[r];

        // ---- oacc += P (16x16) @ V (16x64), V already in registers ----
#pragma unroll
        for (int t = 0; t < 4; ++t) {
            v2f pa;
            pa.x = psh[l][4 * t + kbo];
            pa.y = psh[l][4 * t + kbo + 1];
#pragma unroll
            for (int nt = 0; nt < 4; ++nt)
                oacc[nt] = WMMA4(pa, vf[t][nt], oacc[nt]);
        }
        __syncthreads();
    }

    // ---- normalize and store: out[b, q, h*64 + d] ----
#pragma unroll
    for (int r = 0; r < 8; ++r) {
        const float inv = 1.0f / rsum[r];
        const int m = q0 + r + half * 8;
        const size_t orow = ((size_t)b * Sn + m) * Cn + h * HD;
        out[orow + 0 * 16 + l] = oacc[0][r] * inv;
        out[orow + 1 * 16 + l] = oacc[1][r] * inv;
        out[orow + 2 * 16 + l] = oacc[2][r] * inv;
        out[orow + 3 * 16 + l] = oacc[3][r] * inv;
    }
}

// ---------------------------------------------------------------------------
// Launch. Workspace layout (needs 192 MB):
//   [0,   96MB) : qkv  [4096, 6144] fp32   (later reused for attention output)
//   [96, 128MB) : q packed [B,H,S,64]
//   [128,160MB) : k packed
//   [160,192MB) : v packed
// ---------------------------------------------------------------------------
extern "C" void kernel_launch(void* const* d_in, const int* in_sizes, int n_in,
                              void* d_out, int out_size, void* d_ws, size_t ws_size,
                              hipStream_t stream) {
    (void)in_sizes; (void)n_in; (void)out_size; (void)ws_size;

    const float* x     = (const float*)d_in[0];
    const float* rope  = (const float*)d_in[1];
    const float* w_qkv = (const float*)d_in[2];
    const float* w_out = (const float*)d_in[3];
    const float* qg    = (const float*)d_in[4];
    const float* qb    = (const float*)d_in[5];
    const float* kg    = (const float*)d_in[6];
    const float* kb    = (const float*)d_in[7];
    float* out = (float*)d_out;

    char* ws = (char*)d_ws;
    float* qkv  = (float*)ws;
    float* qpk  = (float*)(ws + (size_t)Mn * NQKV * sizeof(float));
    float* kpk  = qpk + (size_t)Bn * Hn * Sn * HD;
    float* vpk  = kpk + (size_t)Bn * Hn * Sn * HD;
    float* attn = qkv;   // qkv buffer is dead after ln_rope_pack

    // 1) fused QKV projection: [4096,2048] x [6144,2048]^T
    gemm_abt_tdm<<<dim3(Mn / BM, NQKV / BN), 128, 0, stream>>>(x, w_qkv, qkv, Mn, NQKV, Cn);

    // 2) per-head LayerNorm + rotary, repack to [B,H,S,64]
    ln_rope_pack<<<dim3(Bn * Sn * Hn), 32, 0, stream>>>(qkv, rope, qg, qb, kg, kb,
                                                        qpk, kpk, vpk);

    // 3) flash attention (non-causal), output as [B,S,C]
    flash_attn<<<dim3(Sn / 16, Bn * Hn), 32, 0, stream>>>(qpk, kpk, vpk, attn);

    // 4) output projection: [4096,2048] x [2048,2048]^T
    gemm_abt_tdm<<<dim3(Mn / BM, Cn / BN), 128, 0, stream>>>(attn, w_out, out, Mn, Cn, Cn);
}